// Feature_Extration_50044958933192
// MI455X (gfx1250) — compile-verified
//
#include <hip/hip_runtime.h>
#include <hip/hip_bf16.h>

typedef float v2f __attribute__((ext_vector_type(2)));
typedef float v8f __attribute__((ext_vector_type(8)));

static constexpr int NB    = 8;
static constexpr int NPTS  = 2048;
static constexpr int KNN_K = 20;
static constexpr float EPSV = 1e-5f;

// ---------------------------------------------------------------------------
// gfx1250 async global->LDS copy (ASYNCcnt-tracked), with safe fallbacks.
// ---------------------------------------------------------------------------
__device__ __forceinline__ void lds_fill_b32(float* ldst, const float* gsrc) {
#if defined(__gfx1250__) && __has_builtin(__builtin_amdgcn_global_load_async_to_lds_b32)
  float* gs = const_cast<float*>(gsrc);
  __builtin_amdgcn_global_load_async_to_lds_b32(
      (__attribute__((address_space(1))) int*)gs,
      (__attribute__((address_space(3))) int*)ldst, 0, 0);
#else
  *ldst = *gsrc;
#endif
}

__device__ __forceinline__ void async_wait() {
#if defined(__gfx1250__) && __has_builtin(__builtin_amdgcn_s_wait_asynccnt)
  __builtin_amdgcn_s_wait_asynccnt(0);
#elif defined(__gfx1250__)
  asm volatile("s_wait_asynccnt 0x0" ::: "memory");
#endif
}

// ---------------------------------------------------------------------------
// KNN: per-thread query point, async-LDS-tiled candidates, top-20 insert.
// ldB = allocated channel rows per batch (may exceed C due to padding).
// ---------------------------------------------------------------------------
template <int C>
__global__ void knn_kernel(const float* __restrict__ F, int* __restrict__ idx,
                           int ldB, int Npts) {
  const int b = blockIdx.y;
  const int i = blockIdx.x * blockDim.x + threadIdx.x;
  const float* Fb = F + (size_t)b * ldB * Npts;

  float xi[C];
  float xxi = 0.f;
#pragma unroll
  for (int c = 0; c < C; ++c) { xi[c] = Fb[(size_t)c * Npts + i]; xxi += xi[c] * xi[c]; }

  float bd[KNN_K]; int bi[KNN_K];
#pragma unroll
  for (int k = 0; k < KNN_K; ++k) { bd[k] = -3.4e38f; bi[k] = 0; }

  constexpr int TJ = 64;
  __shared__ float Fj[C * TJ];
  __shared__ float xxj[TJ];

  for (int j0 = 0; j0 < Npts; j0 += TJ) {
    __syncthreads();
    for (int e = threadIdx.x; e < C * TJ; e += blockDim.x) {
      int c = e / TJ, j = e % TJ;
      lds_fill_b32(&Fj[e], &Fb[(size_t)c * Npts + j0 + j]);
    }
    async_wait();
    __syncthreads();
    for (int j = threadIdx.x; j < TJ; j += blockDim.x) {
      float s = 0.f;
      for (int c = 0; c < C; ++c) { float v = Fj[c * TJ + j]; s += v * v; }
      xxj[j] = s;
    }
    __syncthreads();
    for (int j = 0; j < TJ; ++j) {
      float inner = 0.f;
#pragma unroll
      for (int c = 0; c < C; ++c) inner += xi[c] * Fj[c * TJ + j];
      float d = 2.f * inner - xxi - xxj[j];
      if (d > bd[KNN_K - 1]) {
        int p = KNN_K - 1;
        while (p > 0 && bd[p - 1] < d) { bd[p] = bd[p - 1]; bi[p] = bi[p - 1]; --p; }
        bd[p] = d; bi[p] = j0 + j;
      }
    }
  }
  int* ob = idx + ((size_t)b * Npts + i) * KNN_K;
  for (int k = 0; k < KNN_K; ++k) ob[k] = bi[k];
}

// ---------------------------------------------------------------------------
// LAFE: tiny per-point attention (3->16). Writes xmanet rows 0..2 = x,
// rows 3..18 = elu(att), row 19 = 0 (K-dim padding for WMMA GEMMs).
// ---------------------------------------------------------------------------
__global__ void lafe_kernel(const float* __restrict__ x, const int* __restrict__ idx,
                            const float* __restrict__ W1, const float* __restrict__ b1,
                            const float* __restrict__ W2, const float* __restrict__ b2,
                            const float* __restrict__ W3, const float* __restrict__ b3,
                            float* __restrict__ xmanet, int Npts) {
  int gid = blockIdx.x * blockDim.x + threadIdx.x;
  if (gid >= NB * Npts) return;
  int b = gid / Npts, n = gid % Npts;
  const float* xb = x + (size_t)b * 3 * Npts;
  float ctr[3];
  for (int c = 0; c < 3; ++c) ctr[c] = xb[(size_t)c * Npts + n];

  float self_att = b3[0];
  for (int o = 0; o < 16; ++o) {
    float nf = b1[o];
    for (int c = 0; c < 3; ++c) nf += W1[o * 3 + c] * ctr[c];
    self_att += W3[o] * nf;
  }

  const int* id = idx + ((size_t)b * Npts + n) * KNN_K;
  float logits[KNN_K];
  for (int k = 0; k < KNN_K; ++k) {
    int j = id[k];
    float xn[3];
    for (int c = 0; c < 3; ++c) xn[c] = ctr[c] - xb[(size_t)c * Npts + j];
    float na = b3[0];
    for (int o = 0; o < 16; ++o) {
      float e = b2[o];
      for (int c = 0; c < 3; ++c) e += W2[o * 3 + c] * xn[c];
      na += W3[o] * e;
    }
    float l = self_att + na;
    logits[k] = l > 0.f ? l : 0.01f * l;
  }
  float mx = -1e30f;
  for (int k = 0; k < KNN_K; ++k) mx = fmaxf(mx, logits[k]);
  float s = 0.f, ew[KNN_K];
  for (int k = 0; k < KNN_K; ++k) { ew[k] = __expf(logits[k] - mx); s += ew[k]; }
  float invs = 1.f / s;

  float vals[16];
  for (int o = 0; o < 16; ++o) vals[o] = 0.f;
  for (int k = 0; k < KNN_K; ++k) {
    int j = id[k];
    float xn[3];
    for (int c = 0; c < 3; ++c) xn[c] = ctr[c] - xb[(size_t)c * Npts + j];
    float w = ew[k] * invs;
    for (int o = 0; o < 16; ++o) {
      float e = b2[o];
      for (int c = 0; c < 3; ++c) e += W2[o * 3 + c] * xn[c];
      vals[o] += w * e;
    }
  }
  float* mb = xmanet + (size_t)b * 20 * Npts;   // padded batch stride
  for (int c = 0; c < 3; ++c) mb[(size_t)c * Npts + n] = ctr[c];
  for (int o = 0; o < 16; ++o) {
    float v = vals[o];
    v = v > 0.f ? v : (__expf(v) - 1.f);
    mb[(size_t)(3 + o) * Npts + n] = v;
  }
  mb[(size_t)19 * Npts + n] = 0.f;              // pad row
}

// ---------------------------------------------------------------------------
// Weight split + zero-pad: wn = W_nb - W_diff, wc = W_ctr + W_diff,
// output stride Cin4 (>= Cin1), pad columns zeroed.
// ---------------------------------------------------------------------------
__global__ void split_w_kernel(const float* __restrict__ Wfull, float* __restrict__ wn,
                               float* __restrict__ wc, int Cout, int Cin1, int Cin4) {
  int e = blockIdx.x * blockDim.x + threadIdx.x;
  if (e >= Cout * Cin4) return;
  int o = e / Cin4, c = e % Cin4;
  float vn = 0.f, vc = 0.f;
  if (c < Cin1) {
    int Ct = 3 * Cin1;
    float nb = Wfull[(size_t)o * Ct + c];
    float cc = Wfull[(size_t)o * Ct + Cin1 + c];
    float df = Wfull[(size_t)o * Ct + 2 * Cin1 + c];
    vn = nb - df;
    vc = cc + df;
  }
  wn[e] = vn;
  wc[e] = vc;
}

// Zero-pad a dense weight matrix (Cout x Cin) -> (Cout x Cin4)
__global__ void pad_w_kernel(const float* __restrict__ W, float* __restrict__ Wp,
                             int Cout, int Cin, int Cin4) {
  int e = blockIdx.x * blockDim.x + threadIdx.x;
  if (e >= Cout * Cin4) return;
  int o = e / Cin4, c = e % Cin4;
  Wp[e] = (c < Cin) ? W[(size_t)o * Cin + c] : 0.f;
}

// ---------------------------------------------------------------------------
// fp32 WMMA GEMM, compile-time K (CIN4 % 4 == 0), branch-free unrolled inner
// loop. B staged via async-to-LDS 16x32 tile shared by the block's 4 waves.
// MODE 1: +bias, bn, relu   MODE 2: bn, leaky(0.2)
// ---------------------------------------------------------------------------
template <int CIN4, int MODE>
__global__ void gemm_wmma_kernel(const float* __restrict__ W, const float* __restrict__ X,
                                 float* __restrict__ Y, int Npts, int ldC, int roff,
                                 const float* __restrict__ bias, const float* __restrict__ g,
                                 const float* __restrict__ bb_, const float* __restrict__ mm,
                                 const float* __restrict__ vv) {
  const int b = blockIdx.z;
  const int t = threadIdx.x;
  const int lane = t & 31, wv = t >> 5;
  const int half = lane >> 4, l15 = lane & 15;
  const int mt = blockIdx.x * 4 + wv;
  const int col0 = blockIdx.y * 16;
  const float* Xb = X + (size_t)b * CIN4 * Npts;
  const int mrow = mt * 16 + l15;
  const float* Wrow = W + (size_t)mrow * CIN4;

  __shared__ __align__(16) float xs[16][32];   // [local col][k in tile]

  v8f acc = {};
  for (int kk0 = 0; kk0 < CIN4; kk0 += 32) {
    const int kt = (CIN4 - kk0 < 32) ? (CIN4 - kk0) : 32;   // multiple of 4
    __syncthreads();
    for (int e = t; e < kt * 16; e += 128) {
      int k = e >> 4, c = e & 15;
      lds_fill_b32(&xs[c][k], &Xb[(size_t)(kk0 + k) * Npts + col0 + c]);
    }
    async_wait();
    __syncthreads();
#pragma unroll
    for (int kk = 0; kk < kt; kk += 4) {
      int ka = kk + 2 * half;
      v2f a  = *(const v2f*)(Wrow + kk0 + ka);
      v2f bv = *(const v2f*)(&xs[l15][ka]);
      acc = __builtin_amdgcn_wmma_f32_16x16x4_f32(false, a, false, bv, (short)0, acc,
                                                  false, false);
    }
  }

  float* Yb = Y + ((size_t)b * ldC + roff) * Npts;
  const int col = col0 + l15;
#pragma unroll
  for (int r = 0; r < 8; ++r) {
    int row = mt * 16 + r + 8 * half;
    float val = acc[r];
    if (MODE == 1) {
      val += bias[row];
      float inv = g[row] * rsqrtf(vv[row] + EPSV);
      val = (val - mm[row]) * inv + bb_[row];
      val = fmaxf(val, 0.f);
    } else if (MODE == 2) {
      float inv = g[row] * rsqrtf(vv[row] + EPSV);
      val = (val - mm[row]) * inv + bb_[row];
      val = val > 0.f ? val : 0.2f * val;
    }
    Yb[(size_t)row * Npts + col] = val;
  }
}

// ---------------------------------------------------------------------------
// Dual-output fp32 WMMA GEMM for the factored edge conv: U = Wn*X, V = Wc*X.
// Shares one staged B tile between both accumulators (raw outputs).
// ---------------------------------------------------------------------------
template <int CIN4>
__global__ void gemm_uv_kernel(const float* __restrict__ Wn, const float* __restrict__ Wc,
                               const float* __restrict__ X, float* __restrict__ U,
                               float* __restrict__ V, int Npts, int Cout) {
  const int b = blockIdx.z;
  const int t = threadIdx.x;
  const int lane = t & 31, wv = t >> 5;
  const int half = lane >> 4, l15 = lane & 15;
  const int mt = blockIdx.x * 4 + wv;
  const int col0 = blockIdx.y * 16;
  const float* Xb = X + (size_t)b * CIN4 * Npts;
  const int mrow = mt * 16 + l15;
  const float* Wnrow = Wn + (size_t)mrow * CIN4;
  const float* Wcrow = Wc + (size_t)mrow * CIN4;

  __shared__ __align__(16) float xs[16][32];

  v8f au = {}, av = {};
  for (int kk0 = 0; kk0 < CIN4; kk0 += 32) {
    const int kt = (CIN4 - kk0 < 32) ? (CIN4 - kk0) : 32;
    __syncthreads();
    for (int e = t; e < kt * 16; e += 128) {
      int k = e >> 4, c = e & 15;
      lds_fill_b32(&xs[c][k], &Xb[(size_t)(kk0 + k) * Npts + col0 + c]);
    }
    async_wait();
    __syncthreads();
#pragma unroll
    for (int kk = 0; kk < kt; kk += 4) {
      int ka = kk + 2 * half;
      v2f an = *(const v2f*)(Wnrow + kk0 + ka);
      v2f ac = *(const v2f*)(Wcrow + kk0 + ka);
      v2f bv = *(const v2f*)(&xs[l15][ka]);
      au = __builtin_amdgcn_wmma_f32_16x16x4_f32(false, an, false, bv, (short)0, au,
                                                 false, false);
      av = __builtin_amdgcn_wmma_f32_16x16x4_f32(false, ac, false, bv, (short)0, av,
                                                 false, false);
    }
  }

  float* Ub = U + (size_t)b * Cout * Npts;
  float* Vb = V + (size_t)b * Cout * Npts;
  const int col = col0 + l15;
#pragma unroll
  for (int r = 0; r < 8; ++r) {
    int row = mt * 16 + r + 8 * half;
    Ub[(size_t)row * Npts + col] = au[r];
    Vb[(size_t)row * Npts + col] = av[r];
  }
}

// ---------------------------------------------------------------------------
// Fused edge-conv epilogue + attention pool. One point per workgroup.
//   x[c,k]  = leaky0.2( bn_conv( U[c, idx[k]] + V[c, n] ) )
//   h       = relu( bn_pool( x ) )          (stored transposed ht[k][c])
//   act     = PW (CxC) * h (Cx20) + pwb      <- WMMA fp32, overlapping col
//             tiles {0..15} and {4..19} so no k-padding / OOB reads needed
//   out[c]  = sum_k x[c,k] * softmax_k(act[c,:])
// ---------------------------------------------------------------------------
template <int C>
__global__ void attpool_kernel(const float* __restrict__ U, const float* __restrict__ V,
                               const int* __restrict__ idx,
                               const float* __restrict__ cg, const float* __restrict__ cb,
                               const float* __restrict__ cm, const float* __restrict__ cv,
                               const float* __restrict__ pg, const float* __restrict__ pb,
                               const float* __restrict__ pm, const float* __restrict__ pv,
                               const float* __restrict__ PW, const float* __restrict__ pwb,
                               const float* __restrict__ copy_src, int copy_rows, int copy_ld,
                               float* __restrict__ out, int ldC, int roff, int zero_row,
                               int Npts) {
  const int n = blockIdx.x, b = blockIdx.y;
  const int t = threadIdx.x;

  __shared__ __align__(16) float xloc[C][KNN_K];
  __shared__ __align__(16) float ht[KNN_K][C];   // transposed: [k][c]
  __shared__ __align__(16) float act[C][KNN_K];
  __shared__ int jloc[KNN_K];

  if (t < KNN_K) jloc[t] = idx[((size_t)b * Npts + n) * KNN_K + t];
  __syncthreads();

  const float* Ub = U + (size_t)b * C * Npts;
  const float* Vb = V + (size_t)b * C * Npts;
  for (int c = t; c < C; c += blockDim.x) {
    float vc    = Vb[(size_t)c * Npts + n];
    float inv_c = cg[c] * rsqrtf(cv[c] + EPSV);
    float inv_p = pg[c] * rsqrtf(pv[c] + EPSV);
    for (int k = 0; k < KNN_K; ++k) {
      float pre = Ub[(size_t)c * Npts + jloc[k]] + vc;
      float xb  = (pre - cm[c]) * inv_c + cb[c];
      float xv  = xb > 0.f ? xb : 0.2f * xb;     // leaky 0.2
      xloc[c][k] = xv;
      float hb  = (xv - pm[c]) * inv_p + pb[c];
      ht[k][c]  = hb > 0.f ? hb : 0.f;           // relu
    }
  }
  __syncthreads();

  const int lane = t & 31, wv = t >> 5;
  const int half = lane >> 4, l15 = lane & 15;
  constexpr int MT = C / 16;
  constexpr int NTT = 2;
  for (int job = wv; job < MT * NTT; job += (int)(blockDim.x >> 5)) {
    int mt = job / NTT, nt = job % NTT;
    int mrow = mt * 16 + l15;
    int col  = (nt == 0 ? 0 : 4) + l15;          // always < KNN_K
    const float* PWrow = PW + (size_t)mrow * C;
    v8f acc = {};
#pragma unroll
    for (int kk = 0; kk < C; kk += 4) {
      int ka = kk + 2 * half;
      v2f a  = *(const v2f*)(PWrow + ka);
      v2f bv = *(const v2f*)(&ht[col][ka]);
      acc = __builtin_amdgcn_wmma_f32_16x16x4_f32(false, a, false, bv, (short)0, acc,
                                                  false, false);
    }
    if (nt == 0 || col >= 16) {                  // write each column exactly once
#pragma unroll
      for (int r = 0; r < 8; ++r) {
        int row = mt * 16 + r + 8 * half;
        act[row][col] = acc[r] + pwb[row];
      }
    }
  }
  __syncthreads();

  for (int c = t; c < C; c += blockDim.x) {
    float mx = -1e30f;
    for (int k = 0; k < KNN_K; ++k) mx = fmaxf(mx, act[c][k]);
    float s = 0.f, ew[KNN_K];
    for (int k = 0; k < KNN_K; ++k) { ew[k] = __expf(act[c][k] - mx); s += ew[k]; }
    float invs = 1.f / s;
    float o = 0.f;
    for (int k = 0; k < KNN_K; ++k) o += xloc[c][k] * ew[k] * invs;
    out[(((size_t)b * ldC) + roff + c) * Npts + n] = o;
  }
  for (int c = t; c < copy_rows; c += blockDim.x)
    out[(((size_t)b * ldC) + c) * Npts + n] =
        copy_src[((size_t)b * copy_ld + c) * Npts + n];
  if (zero_row >= 0 && t == 0)
    out[(((size_t)b * ldC) + zero_row) * Npts + n] = 0.f;
}

// ---------------------------------------------------------------------------
static inline char* carve(char*& p, size_t bytes) {
  char* r = p;
  p += (bytes + 255) & ~size_t(255);
  return r;
}

extern "C" void kernel_launch(void* const* d_in, const int* in_sizes, int n_in,
                              void* d_out, int out_size, void* d_ws, size_t ws_size,
                              hipStream_t stream) {
  const float* x     = (const float*)d_in[0];
  const float* law1  = (const float*)d_in[1];
  const float* lab1  = (const float*)d_in[2];
  const float* law2  = (const float*)d_in[3];
  const float* lab2  = (const float*)d_in[4];
  const float* law3  = (const float*)d_in[5];
  const float* lab3  = (const float*)d_in[6];
  const float* mlp_w = (const float*)d_in[7];
  const float* mlp_b = (const float*)d_in[8];
  const float* mlp_g = (const float*)d_in[9];
  const float* mlp_be= (const float*)d_in[10];
  const float* mlp_m = (const float*)d_in[11];
  const float* mlp_v = (const float*)d_in[12];
  const float* c1_w  = (const float*)d_in[13];
  const float* bn1_g = (const float*)d_in[14];
  const float* bn1_b = (const float*)d_in[15];
  const float* bn1_m = (const float*)d_in[16];
  const float* bn1_v = (const float*)d_in[17];
  const float* p1_g  = (const float*)d_in[18];
  const float* p1_b  = (const float*)d_in[19];
  const float* p1_m  = (const float*)d_in[20];
  const float* p1_v  = (const float*)d_in[21];
  const float* p1_w  = (const float*)d_in[22];
  const float* p1_wb = (const float*)d_in[23];
  const float* c2_w  = (const float*)d_in[24];
  const float* bn2_g = (const float*)d_in[25];
  const float* bn2_b = (const float*)d_in[26];
  const float* bn2_m = (const float*)d_in[27];
  const float* bn2_v = (const float*)d_in[28];
  const float* p2_g  = (const float*)d_in[29];
  const float* p2_b  = (const float*)d_in[30];
  const float* p2_m  = (const float*)d_in[31];
  const float* p2_v  = (const float*)d_in[32];
  const float* p2_w  = (const float*)d_in[33];
  const float* p2_wb = (const float*)d_in[34];
  const float* c3_w  = (const float*)d_in[35];
  const float* bn3_g = (const float*)d_in[36];
  const float* bn3_b = (const float*)d_in[37];
  const float* bn3_m = (const float*)d_in[38];
  const float* bn3_v = (const float*)d_in[39];

  char* wp = (char*)d_ws;
  float* xmanet = (float*)carve(wp, (size_t)NB * 20 * NPTS * 4);   // 19 + pad row
  int*   idx0   = (int*)  carve(wp, (size_t)NB * NPTS * KNN_K * 4);
  int*   idx1   = (int*)  carve(wp, (size_t)NB * NPTS * KNN_K * 4);
  int*   idx2   = (int*)  carve(wp, (size_t)NB * NPTS * KNN_K * 4);
  float* xc1    = (float*)carve(wp, (size_t)NB * 84 * NPTS * 4);   // 83 + pad row
  float* xc2    = (float*)carve(wp, (size_t)NB * 512 * NPTS * 4);
  float* U1     = (float*)carve(wp, (size_t)NB * 64 * NPTS * 4);
  float* V1     = (float*)carve(wp, (size_t)NB * 64 * NPTS * 4);
  float* U2     = (float*)carve(wp, (size_t)NB * 256 * NPTS * 4);
  float* V2     = (float*)carve(wp, (size_t)NB * 256 * NPTS * 4);
  float* wn1    = (float*)carve(wp, (size_t)64 * 20 * 4);
  float* wc1    = (float*)carve(wp, (size_t)64 * 20 * 4);
  float* wn2    = (float*)carve(wp, (size_t)256 * 84 * 4);
  float* wc2    = (float*)carve(wp, (size_t)256 * 84 * 4);
  float* mlp_wp = (float*)carve(wp, (size_t)256 * 20 * 4);
  float* outp   = (float*)d_out;

  dim3 knn_grid(NPTS / 256, NB);

  // 1) knn on raw xyz, then LAFE -> xmanet (padded to 20 rows)
  knn_kernel<3><<<knn_grid, 256, 0, stream>>>(x, idx0, 3, NPTS);
  lafe_kernel<<<(NB * NPTS + 255) / 256, 256, 0, stream>>>(
      x, idx0, law1, lab1, law2, lab2, law3, lab3, xmanet, NPTS);

  // 2) weight prep: split + zero-pad to WMMA-friendly K
  split_w_kernel<<<(64 * 20 + 255) / 256, 256, 0, stream>>>(c1_w, wn1, wc1, 64, 19, 20);
  split_w_kernel<<<(256 * 84 + 255) / 256, 256, 0, stream>>>(c2_w, wn2, wc2, 256, 83, 84);
  pad_w_kernel<<<(256 * 20 + 255) / 256, 256, 0, stream>>>(mlp_w, mlp_wp, 256, 19, 20);

  // 3) mlp branch: xc2 rows 0..255 = relu(bn(mlp_w @ xmanet + mlp_b))
  gemm_wmma_kernel<20, 1><<<dim3(4, NPTS / 16, NB), 128, 0, stream>>>(
      mlp_wp, xmanet, xc2, NPTS, 512, 0, mlp_b, mlp_g, mlp_be, mlp_m, mlp_v);

  // 4) block 1: knn(19ch), fused U1/V1 GEMM, fused conv-bn-leaky + att pool
  knn_kernel<19><<<knn_grid, 256, 0, stream>>>(xmanet, idx1, 20, NPTS);
  gemm_uv_kernel<20><<<dim3(1, NPTS / 16, NB), 128, 0, stream>>>(
      wn1, wc1, xmanet, U1, V1, NPTS, 64);
  attpool_kernel<64><<<dim3(NPTS, NB), 256, 0, stream>>>(
      U1, V1, idx1, bn1_g, bn1_b, bn1_m, bn1_v, p1_g, p1_b, p1_m, p1_v,
      p1_w, p1_wb, xmanet, 19, 20, xc1, 84, 19, 83, NPTS);

  // 5) block 2: knn(83ch), fused U2/V2 GEMM, fused att pool -> xc2 rows 256..511
  knn_kernel<83><<<knn_grid, 256, 0, stream>>>(xc1, idx2, 84, NPTS);
  gemm_uv_kernel<84><<<dim3(4, NPTS / 16, NB), 128, 0, stream>>>(
      wn2, wc2, xc1, U2, V2, NPTS, 256);
  attpool_kernel<256><<<dim3(NPTS, NB), 256, 0, stream>>>(
      U2, V2, idx2, bn2_g, bn2_b, bn2_m, bn2_v, p2_g, p2_b, p2_m, p2_v,
      p2_w, p2_wb, nullptr, 0, 0, xc2, 512, 256, -1, NPTS);

  // 6) final 512x512 conv + bn + leaky(0.2) -> d_out
  gemm_wmma_kernel<512, 2><<<dim3(8, NPTS / 16, NB), 128, 0, stream>>>(
      c3_w, xc2, outp, NPTS, 512, 0, nullptr, bn3_g, bn3_b, bn3_m, bn3_v);
}